// interactionModule_9517647528245
// MI455X (gfx1250) — compile-verified
//
#include <hip/hip_runtime.h>

// Physics constants from the reference
#define GAMMA_F 0.1f
#define MIN_R_F 0.1f
#define EPS_F   1e-12f

constexpr int CHUNK = 2048;        // edges staged per block-iteration
constexpr int TPB   = 256;         // 8 waves (wave32)
constexpr int EPT   = CHUNK / TPB; // 8 edges per thread (32B per array)

// ---------------------------------------------------------------------------
// Kernel 1: acc = -GAMMA * v  (elementwise over 2*N floats). Also re-inits
// d_out every call (harness poisons once, never restores between replays).
// ---------------------------------------------------------------------------
__global__ void init_acc_kernel(const float* __restrict__ v,
                                float* __restrict__ out, int n2) {
    int i = blockIdx.x * blockDim.x + threadIdx.x;
    if (i < n2) out[i] = -GAMMA_F * v[i];
}

// ---------------------------------------------------------------------------
// Kernel 2: per-edge LJ force + scatter-add.
// Edge indices stream HBM->LDS via gfx1250 async copies (ASYNCcnt), double
// buffered: the next 16KB chunk is in flight while the current one computes.
// Position gathers hit L2/WGP$ (x is 800KB << 192MB L2). Scatter uses the
// native no-return global_atomic_add_f32 (no CAS fallback possible).
// All index math is int32 (E*4 = 25.6MB < 2^31) to cut VALU/SALU overhead.
// ---------------------------------------------------------------------------
__global__ __launch_bounds__(TPB) void edge_force_kernel(
    const float* __restrict__ x,
    const int*  __restrict__ esrc,
    const int*  __restrict__ edst,
    float* __restrict__ out, int E, int gridBlocks) {

    // Single shared array -> LDS byte offset 0; layout:
    //   [buf0 src | buf1 src | buf0 dst | buf1 dst], each CHUNK ints.
    __shared__ int smem[4 * CHUNK];

    // Escape the LDS pointer so the compiler must honor reads of smem that
    // the async-copy asm produced (it cannot see those writes).
    {
        int* sp = smem;
        asm volatile("" : "+v"(sp) :: "memory");
    }

    const int t = threadIdx.x;
    const int nChunks = (E + CHUNK - 1) / CHUNK;
    int c = blockIdx.x;
    if (c >= nChunks) return;

    const float2* __restrict__ xp = reinterpret_cast<const float2*>(x);

    // Stage one chunk's src+dst indices: per thread 32B per array = 2x b128.
    // GVS addressing: mem = SGPR64(base) + VGPR32(byteOff)+inst_offset; the
    // first VGPR operand is the LDS destination byte offset.
    auto issue = [&](int chunk, int buf) {
        int gi   = chunk * CHUNK + t * EPT;      // first edge staged by thread
        int last = E - EPT;                      // clamp tail (OOB-safe)
        gi = gi > last ? last : gi;
        gi = gi < 0 ? 0 : gi;
        unsigned byteOff = (unsigned)gi * 4u;
        unsigned ldsSrc  = (unsigned)((buf * CHUNK + t * EPT) * 4);
        unsigned ldsDst  = (unsigned)(((2 + buf) * CHUNK + t * EPT) * 4);
        asm volatile("global_load_async_to_lds_b128 %0, %1, %2"
                     :: "v"(ldsSrc), "v"(byteOff), "s"(esrc) : "memory");
        asm volatile("global_load_async_to_lds_b128 %0, %1, %2 offset:16"
                     :: "v"(ldsSrc), "v"(byteOff), "s"(esrc) : "memory");
        asm volatile("global_load_async_to_lds_b128 %0, %1, %2"
                     :: "v"(ldsDst), "v"(byteOff), "s"(edst) : "memory");
        asm volatile("global_load_async_to_lds_b128 %0, %1, %2 offset:16"
                     :: "v"(ldsDst), "v"(byteOff), "s"(edst) : "memory");
    };

    issue(c, 0);
    int buf = 0;

    for (; c < nChunks; c += gridBlocks) {
        int nxt = c + gridBlocks;
        bool prefetch = (nxt < nChunks);          // block-uniform branch
        if (prefetch) {
            issue(nxt, buf ^ 1);
            // Current chunk = oldest 4 async ops; per-wave in-order
            // completion => ASYNCcnt<=4 means they landed while the next
            // chunk's 4 ops stay in flight.
            asm volatile("s_wait_asynccnt 4" ::: "memory");
        } else {
            asm volatile("s_wait_asynccnt 0" ::: "memory");
        }

        const int ebase = c * CHUNK + t * EPT;
        const int lbase = t * EPT;
#pragma unroll
        for (int k = 0; k < EPT; ++k) {
            int e = ebase + k;
            if (e < E) {
                int s = smem[buf * CHUNK + lbase + k];
                int d = smem[(2 + buf) * CHUNK + lbase + k];
                float2 xs = xp[s];
                float2 xd = xp[d];
                float dx = xd.x - xs.x;
                float dy = xd.y - xs.y;
                float rin = sqrtf(dx * dx + dy * dy);    // |dr|
                float r   = fmaxf(rin, MIN_R_F);         // distance clamp
                float inv = 1.0f / r;                    // s = R_C/r, R_C=1
                float inv2 = inv * inv;
                float inv6 = inv2 * inv2 * inv2;         // s^6
                float F = 4.0f * inv6 * (12.0f * inv6 - 6.0f) * inv;
                float scale = F / fmaxf(rin, EPS_F);     // F * normalize(dr)
                float mx = scale * dx;
                float my = scale * dy;
                unsigned long long addr = (unsigned long long)(out + 2 * d);
                // native no-return f32 atomics (STOREcnt; implicit wait-idle
                // at s_endpgm). volatile pins them; no memory clobber needed
                // since nothing in this kernel reads `out`.
                asm volatile("global_atomic_add_f32 %0, %1, off"
                             :: "v"(addr), "v"(mx));
                asm volatile("global_atomic_add_f32 %0, %1, off offset:4"
                             :: "v"(addr), "v"(my));
            }
        }
        buf ^= 1;
    }
}

// ---------------------------------------------------------------------------
// inputs: 0=x [N,2] f32, 1=v [N,2] f32, 2=edge_src [E] int, 3=edge_dst [E] int
// output: acc [N,2] f32
// ---------------------------------------------------------------------------
extern "C" void kernel_launch(void* const* d_in, const int* in_sizes, int n_in,
                              void* d_out, int out_size, void* d_ws, size_t ws_size,
                              hipStream_t stream) {
    const float* x    = (const float*)d_in[0];
    const float* v    = (const float*)d_in[1];
    const int*   esrc = (const int*)d_in[2];
    const int*   edst = (const int*)d_in[3];
    float* out = (float*)d_out;

    const int n2 = in_sizes[1];  // 2*N_NODES (flat v)
    const int E  = in_sizes[2];  // N_EDGES

    init_acc_kernel<<<(n2 + 255) / 256, 256, 0, stream>>>(v, out, n2);

    int nChunks = (E + CHUNK - 1) / CHUNK;
    int blocks = nChunks < 1024 ? nChunks : 1024;  // persistent blocks; ~3
    if (blocks > 0) {                              // chunks each => prefetch
        edge_force_kernel<<<blocks, TPB, 0, stream>>>(x, esrc, edst, out, E,
                                                      blocks);
    }
}